// INSTNCT_88613765251433
// MI455X (gfx1250) — compile-verified
//
#include <hip/hip_runtime.h>

#define B_ 32
#define M_ 32768
#define D_ 128
#define W_ 129   // 2*R+1, R=64

typedef __attribute__((ext_vector_type(2))) float v2f;
typedef __attribute__((ext_vector_type(4))) float v4f;
typedef __attribute__((ext_vector_type(8))) float v8f;

// ---------------------------------------------------------------------------
// Kernel 1: bulk streaming copy ring -> out.
// 512 MB in + 512 MB out, pure bandwidth. b128 accesses + non-temporal TH
// (data is streamed once; 512 MB >> 192 MB L2, so NT avoids cache pollution).
// Unroll x4 across the grid stride: 4 independent global_load_b128 in flight
// per thread, stores coalesced across lanes.
// ---------------------------------------------------------------------------
__global__ __launch_bounds__(256) void ring_copy(const v4f* __restrict__ src,
                                                 v4f* __restrict__ dst,
                                                 size_t n4) {
  const size_t stride = (size_t)gridDim.x * blockDim.x;
  size_t i = (size_t)blockIdx.x * blockDim.x + threadIdx.x;
  for (; i + 3 * stride < n4; i += 4 * stride) {
    v4f a = __builtin_nontemporal_load(src + i);
    v4f b = __builtin_nontemporal_load(src + i + stride);
    v4f c = __builtin_nontemporal_load(src + i + 2 * stride);
    v4f d = __builtin_nontemporal_load(src + i + 3 * stride);
    __builtin_nontemporal_store(a, dst + i);
    __builtin_nontemporal_store(b, dst + i + stride);
    __builtin_nontemporal_store(c, dst + i + 2 * stride);
    __builtin_nontemporal_store(d, dst + i + 3 * stride);
  }
  for (; i < n4; i += stride) {
    __builtin_nontemporal_store(__builtin_nontemporal_load(src + i), dst + i);
  }
}

// ---------------------------------------------------------------------------
// Kernel 2: windowed update via WMMA rank-1 outer product.
//   updated = current*(1 - e*w) + (g*w) (x) write_vec
// One wave32 per 16x16 (rows=window positions, cols=D chunk) tile:
//   A (16x4 f32):  A[m][0] = g*w[m], A[m][1..3] = 0
//   B (4x16 f32):  B[0][n] = write_vec[n], B[1..3][n] = 0
//   C (16x16 f32): current[m][n] * (1 - e*w[m])   (0 for padded rows)
//   D = A*B + C  -> exactly the updated tile.
// Lane layout per CDNA5 ISA 7.12.2:
//   A: lane L holds M=L&15, regs {K = r + 2*(L>>4)}  -> only lanes 0..15, reg0
//   B: lane L holds N=L&15, regs {K = r + 2*(L>>4)}  -> only lanes 0..15, reg0
//   C/D: lane L, reg r holds M = r + 8*(L>>4), N = L&15
// All pre-WMMA masking uses selects (EXEC stays all-ones as WMMA requires);
// row-validity branching happens only on the post-WMMA scatter stores.
// ---------------------------------------------------------------------------
__global__ __launch_bounds__(256) void ring_window_update(
    const float* __restrict__ ring, const float* __restrict__ wv,
    const float* __restrict__ w, const float* __restrict__ erase,
    const float* __restrict__ gate, const int* __restrict__ idx,
    float* __restrict__ out) {
  constexpr int ROW_TILES = (W_ + 15) / 16;  // 9
  constexpr int COL_TILES = D_ / 16;         // 8
  constexpr int TILES = B_ * ROW_TILES * COL_TILES;  // 2304

  const int wave = (int)((blockIdx.x * blockDim.x + threadIdx.x) >> 5);
  const int lane = (int)(threadIdx.x & 31);
  if (wave >= TILES) return;  // whole wave exits together

  const int b   = wave / (ROW_TILES * COL_TILES);
  const int rem = wave % (ROW_TILES * COL_TILES);
  const int rt  = rem / COL_TILES;
  const int ct  = rem % COL_TILES;

  const int half = lane >> 4;   // 0: lanes 0-15, 1: lanes 16-31
  const int n    = lane & 15;
  const int col  = ct * 16 + n; // < 128 always

  const float e = erase[b];
  const float g = gate[b];

  // A-matrix: rank-1 column (K=0 only). Lanes 16-31 carry K=2,3 -> zero.
  const int mA = rt * 16 + n;
  const float aval = (half == 0 && mA < W_) ? g * w[b * W_ + mA] : 0.0f;
  v2f A;
  A.x = aval;
  A.y = 0.0f;

  // B-matrix: rank-1 row (K=0 only).
  const float bval = (half == 0) ? wv[b * D_ + col] : 0.0f;
  v2f Bm;
  Bm.x = bval;
  Bm.y = 0.0f;

  // C-matrix: current * (1 - e*w) for this lane's 8 rows.
  v8f C;
#pragma unroll
  for (int r = 0; r < 8; ++r) {
    const int m  = rt * 16 + r + 8 * half;
    const int mc = (m < W_) ? m : (W_ - 1);          // clamp, select not branch
    const int row = idx[b * W_ + mc];
    const float cur = ring[((size_t)b * M_ + row) * D_ + col];
    const float wm  = w[b * W_ + mc];
    C[r] = (m < W_) ? cur * (1.0f - e * wm) : 0.0f;
  }

  // D = A x B + C  (v_wmma_f32_16x16x4_f32)
  v8f Dm = __builtin_amdgcn_wmma_f32_16x16x4_f32(
      /*neg_a=*/false, A, /*neg_b=*/false, Bm,
      /*c_mod=*/(short)0, C, /*reuse_a=*/false, /*reuse_b=*/false);

  // Scatter the updated tile back over the copied ring.
#pragma unroll
  for (int r = 0; r < 8; ++r) {
    const int m = rt * 16 + r + 8 * half;
    if (m < W_) {
      const int row = idx[b * W_ + m];
      out[((size_t)b * M_ + row) * D_ + col] = Dm[r];
    }
  }
}

extern "C" void kernel_launch(void* const* d_in, const int* in_sizes, int n_in,
                              void* d_out, int out_size, void* d_ws, size_t ws_size,
                              hipStream_t stream) {
  const float* ring  = (const float*)d_in[0];  // (B, M, D)
  const float* wv    = (const float*)d_in[1];  // (B, D)
  const float* w     = (const float*)d_in[2];  // (B, W)
  const float* erase = (const float*)d_in[3];  // (B,)
  const float* gate  = (const float*)d_in[4];  // (B,)
  const int*   idx   = (const int*)d_in[5];    // (B, W)
  float* out = (float*)d_out;                  // (B, M, D)

  // Bulk copy: 33,554,432 float4s; 8192 blocks x 256 thr -> 16 float4/thread.
  const size_t n4 = (size_t)B_ * M_ * D_ / 4;
  ring_copy<<<8192, 256, 0, stream>>>((const v4f*)ring, (v4f*)out, n4);

  // Window update: 2304 tile-waves, 8 waves/block -> 288 blocks.
  constexpr int TILES = B_ * ((W_ + 15) / 16) * (D_ / 16);
  const int blocks = (TILES * 32 + 255) / 256;
  ring_window_update<<<blocks, 256, 0, stream>>>(ring, wv, w, erase, gate, idx, out);
}